// DecorrLoss_41618233099088
// MI455X (gfx1250) — compile-verified
//
#include <hip/hip_runtime.h>
#include <cstdint>

typedef __attribute__((ext_vector_type(2))) float v2f;
typedef __attribute__((ext_vector_type(8))) float v8f;

#define D        1024
#define MTOT     32768
#define KAPPA    0.5f
#define BT       128              // output tile edge per workgroup
#define KT       32               // K-slab rows staged in LDS
#define KSPLIT   8                // split-K factor (grid.z)
#define MCHUNK   (MTOT / KSPLIT)  // 4096 rows per workgroup
#define LSTRIDE  144              // 128 + 16 pad: 2*144 % 64 == 32 -> lane halves hit disjoint banks
#define SLAB     (KT * LSTRIDE)   // floats per LDS slab

__global__ __launch_bounds__(256) void zero_kernel(float* __restrict__ out, int n) {
    int i = blockIdx.x * 256 + threadIdx.x;
    if (i < n) out[i] = 0.0f;
}

// Async copy of 16 bytes (per lane) global -> LDS, tracked by ASYNCcnt.
__device__ __forceinline__ void async_b128_to_lds(uint32_t ldsAddr, int gByteOff,
                                                  const float* base) {
    asm volatile("global_load_async_to_lds_b128 %0, %1, %2"
                 :: "v"(ldsAddr), "v"(gByteOff), "s"(base)
                 : "memory");
}

// Issue one KT x 128 slab pair (A-panel cols i0, B-panel cols j0) into LDS buffers.
__device__ __forceinline__ void issue_slab(const float* __restrict__ X, int m0,
                                           int i0, int j0,
                                           float* bufA, float* bufB,
                                           int ldRow, int ldCol) {
    #pragma unroll
    for (int q = 0; q < 4; ++q) {
        const int r     = ldRow + (q << 3);
        const int rowB  = (m0 + r) * D;
        const uint32_t la = (uint32_t)(uintptr_t)(bufA + r * LSTRIDE + ldCol);
        const uint32_t lb = (uint32_t)(uintptr_t)(bufB + r * LSTRIDE + ldCol);
        async_b128_to_lds(la, (rowB + i0 + ldCol) * 4, X);
        async_b128_to_lds(lb, (rowB + j0 + ldCol) * 4, X);
    }
}

// S += X[mchunk]^T * X[mchunk] for one 128x128 tile, via V_WMMA_F32_16X16X4_F32,
// double-buffered async global->LDS staging.
__global__ __launch_bounds__(256) void xtx_kernel(const float* __restrict__ X,
                                                  float* __restrict__ out) {
    __shared__ float smem[4 * SLAB];  // [buf0:A,B][buf1:A,B]

    const int i0    = blockIdx.x * BT;
    const int j0    = blockIdx.y * BT;
    const int mbase = blockIdx.z * MCHUNK;

    const int tid  = threadIdx.x;
    const int lane = tid & 31;
    const int wave = tid >> 5;          // 0..7
    const int wi   = (wave >> 2) * 64;  // wave grid 2 (i) x 4 (j)
    const int wj   = (wave & 3) * 32;

    v8f acc[4][2];
    #pragma unroll
    for (int a = 0; a < 4; ++a)
        #pragma unroll
        for (int b = 0; b < 2; ++b)
            acc[a][b] = (v8f){};

    // fragment addressing (32-bit A 16x4 layout; B mirrors it):
    // lanes 0-15 supply K=0,1 ; lanes 16-31 supply K=2,3 ; column = lane&15
    const int fr = (lane >> 4) << 1;    // 0 or 2
    const int fc = lane & 15;

    // LDS fill mapping: wave w covers rows w, w+8, w+16, w+24; lane covers 4 floats
    const int ldRow = wave;
    const int ldCol = lane << 2;

    // prologue: stage slab 0 into buffer 0
    issue_slab(X, mbase, i0, j0, &smem[0], &smem[SLAB], ldRow, ldCol);

    int cur = 0;
    for (int mo = 0; mo < MCHUNK; mo += KT) {
        const int nxt = cur ^ 1;
        const bool hasNext = (mo + KT) < MCHUNK;
        if (hasNext) {
            // stage slab s+1 into the other buffer while slab s finishes
            issue_slab(X, mbase + mo + KT, i0, j0,
                       &smem[(nxt << 1) * SLAB], &smem[((nxt << 1) + 1) * SLAB],
                       ldRow, ldCol);
            asm volatile("s_wait_asynccnt 0x8" ::: "memory");  // slab s resident
        } else {
            asm volatile("s_wait_asynccnt 0x0" ::: "memory");
        }
        __syncthreads();   // all waves' async fills for slab s visible

        const float* curA = &smem[(cur << 1) * SLAB];
        const float* curB = &smem[((cur << 1) + 1) * SLAB];

        #pragma unroll
        for (int kk = 0; kk < KT; kk += 4) {
            v2f af[4], bfr[2];
            #pragma unroll
            for (int a = 0; a < 4; ++a) {
                const int c = wi + a * 16 + fc;
                af[a].x = curA[(kk + fr)     * LSTRIDE + c];
                af[a].y = curA[(kk + fr + 1) * LSTRIDE + c];
            }
            #pragma unroll
            for (int b = 0; b < 2; ++b) {
                const int c = wj + b * 16 + fc;
                bfr[b].x = curB[(kk + fr)     * LSTRIDE + c];
                bfr[b].y = curB[(kk + fr + 1) * LSTRIDE + c];
            }
            #pragma unroll
            for (int a = 0; a < 4; ++a)
                #pragma unroll
                for (int b = 0; b < 2; ++b)
                    acc[a][b] = __builtin_amdgcn_wmma_f32_16x16x4_f32(
                        false, af[a], false, bfr[b],
                        (short)0, acc[a][b], false, false);
        }
        __syncthreads();   // everyone done reading buf cur before it is refilled
        cur = nxt;
    }

    // C/D layout: VGPR v -> row v (lanes 0-15) / v+8 (lanes 16-31); col = lane&15
    const int rofs = (lane >> 4) << 3;
    #pragma unroll
    for (int a = 0; a < 4; ++a) {
        #pragma unroll
        for (int b = 0; b < 2; ++b) {
            const int col = j0 + wj + b * 16 + fc;
            #pragma unroll
            for (int v = 0; v < 8; ++v) {
                const int row = i0 + wi + a * 16 + v + rofs;
                atomicAdd(&out[(size_t)row * D + col], acc[a][b][v]);
            }
        }
    }
}

// Per-row: ss = sum x^2, s4 = sum x^4, wh = sum (x^2-1)^2. One wave per row.
__global__ __launch_bounds__(256) void loss_kernel(const float* __restrict__ X,
                                                   float* __restrict__ out) {
    const int lane = threadIdx.x & 31;
    const int wave = threadIdx.x >> 5;
    const int row  = blockIdx.x * 8 + wave;
    const float4* xr = (const float4*)(X + (size_t)row * D);

    float ss = 0.0f, s4 = 0.0f, wh = 0.0f;
    #pragma unroll
    for (int q = 0; q < 8; ++q) {
        float4 v = xr[lane + (q << 5)];
        float t, u;
        t = v.x * v.x; ss += t; s4 += t * t; u = t - 1.0f; wh += u * u;
        t = v.y * v.y; ss += t; s4 += t * t; u = t - 1.0f; wh += u * u;
        t = v.z * v.z; ss += t; s4 += t * t; u = t - 1.0f; wh += u * u;
        t = v.w * v.w; ss += t; s4 += t * t; u = t - 1.0f; wh += u * u;
    }
    #pragma unroll
    for (int off = 16; off > 0; off >>= 1) {
        ss += __shfl_xor(ss, off);
        s4 += __shfl_xor(s4, off);
        wh += __shfl_xor(wh, off);
    }
    if (lane == 0) {
        atomicAdd(&out[(size_t)D * D],     ss * ss - s4);
        atomicAdd(&out[(size_t)D * D + 1], wh);
    }
}

// grad[i][j] = (1-k)*S[ij]/M off-diag; k*(S[ii]/M - 1) on diag; scale scalars.
__global__ __launch_bounds__(256) void finalize_kernel(float* __restrict__ out) {
    const size_t idx  = (size_t)blockIdx.x * 256 + threadIdx.x;
    const float  invM = 1.0f / (float)MTOT;
    if (idx < (size_t)D * D) {
        const int row = (int)(idx >> 10);
        const int col = (int)(idx & (D - 1));
        const float s = out[idx] * invM;
        out[idx] = (row == col) ? (KAPPA * (s - 1.0f)) : ((1.0f - KAPPA) * s);
    }
    if (idx == 0) {
        out[(size_t)D * D]     *= 1.0f / ((float)MTOT * (float)D * (float)D);
        out[(size_t)D * D + 1] *= 1.0f / ((float)MTOT * (float)D);
    }
}

extern "C" void kernel_launch(void* const* d_in, const int* in_sizes, int n_in,
                              void* d_out, int out_size, void* d_ws, size_t ws_size,
                              hipStream_t stream) {
    (void)in_sizes; (void)n_in; (void)d_ws; (void)ws_size; (void)out_size;
    const float* X  = (const float*)d_in[0];
    float* out      = (float*)d_out;

    const int ntot = D * D + 2;
    zero_kernel<<<(ntot + 255) / 256, 256, 0, stream>>>(out, ntot);

    dim3 grid(D / BT, D / BT, KSPLIT);
    xtx_kernel<<<grid, 256, 0, stream>>>(X, out);

    loss_kernel<<<MTOT / 8, 256, 0, stream>>>(X, out);

    finalize_kernel<<<(D * D + 255) / 256, 256, 0, stream>>>(out);
}